// BilinearChebConv_3255585210961
// MI455X (gfx1250) — compile-verified
//
#include <hip/hip_runtime.h>

typedef __attribute__((ext_vector_type(2))) float v2f;
typedef __attribute__((ext_vector_type(8))) float v8f;

#define MDIM 1536
#define NDIM 1536
#define NP   5      // P+1 (Chebyshev order + 1)
#define OUTC 32

static __device__ __forceinline__ v8f wmma_f32(v2f a, v2f b, v8f c) {
    return __builtin_amdgcn_wmma_f32_16x16x4_f32(false, a, false, b, (short)0, c,
                                                 false, false);
}

// ---------------------------------------------------------------------------
// T[0] = I, T[1] = L   (first two Chebyshev polynomials)
// ---------------------------------------------------------------------------
__global__ __launch_bounds__(256) void init_cheb01(const float* __restrict__ L,
                                                   float* __restrict__ T, int n) {
    size_t idx = (size_t)blockIdx.x * blockDim.x + threadIdx.x;
    size_t total = (size_t)n * n;
    if (idx >= total) return;
    int r = (int)(idx / n), c = (int)(idx % n);
    T[idx]         = (r == c) ? 1.0f : 0.0f;
    T[total + idx] = L[idx];
}

__global__ __launch_bounds__(256) void copy_f32(const float* __restrict__ s,
                                                float* __restrict__ d, size_t n) {
    size_t idx = (size_t)blockIdx.x * blockDim.x + threadIdx.x;
    if (idx < n) d[idx] = s[idx];
}

// ---------------------------------------------------------------------------
// D = alpha * A(MxK) @ B(KxN) + beta * Cin     (f32 WMMA, wave = 32x32 block)
// gridDim.z batches independent GEMMs via sAz/sBz/sDz element strides.
// ---------------------------------------------------------------------------
__global__ __launch_bounds__(256)
void gemm32_f32(const float* __restrict__ A, const float* __restrict__ B,
                const float* __restrict__ Cin, float* __restrict__ D,
                int Md, int Nd, int Kd, float alpha, float beta,
                long long sAz, long long sBz, long long sDz) {
    A += (long long)blockIdx.z * sAz;
    B += (long long)blockIdx.z * sBz;
    D += (long long)blockIdx.z * sDz;

    const int lane   = threadIdx.x & 31;
    const int wave   = blockIdx.x * (blockDim.x >> 5) + (threadIdx.x >> 5);
    const int tilesN = Nd >> 5;
    const int tm     = (wave / tilesN) << 5;  // 32-row block base
    const int tn     = (wave % tilesN) << 5;  // 32-col block base
    if (tm >= Md) return;

    const int mr = lane & 15;          // row/col within 16-tile
    const int kb = (lane >> 4) << 1;   // 0 or 2: K sub-pair per lane half

    v8f acc[2][2] = {};

    const float* a0 = A + (size_t)(tm + mr) * Kd + kb;
    const float* a1 = a0 + (size_t)16 * Kd;
    const float* b0 = B + (size_t)kb * Nd + tn + mr;

    for (int k = 0; k < Kd; k += 4) {
        v2f af0 = *(const v2f*)(a0 + k);
        v2f af1 = *(const v2f*)(a1 + k);
        const float* bk = b0 + (size_t)k * Nd;
        v2f bf0, bf1;
        bf0.x = bk[0];  bf0.y = bk[Nd];
        bf1.x = bk[16]; bf1.y = bk[16 + Nd];
        acc[0][0] = wmma_f32(af0, bf0, acc[0][0]);
        acc[0][1] = wmma_f32(af0, bf1, acc[0][1]);
        acc[1][0] = wmma_f32(af1, bf0, acc[1][0]);
        acc[1][1] = wmma_f32(af1, bf1, acc[1][1]);
    }

    const int rb = (lane >> 4) << 3;   // +8 rows for upper lane half
#pragma unroll
    for (int i = 0; i < 2; ++i)
#pragma unroll
        for (int j = 0; j < 2; ++j) {
            const int row0 = tm + i * 16 + rb;
            const int col  = tn + j * 16 + mr;
#pragma unroll
            for (int r = 0; r < 8; ++r) {
                size_t idx = (size_t)(row0 + r) * Nd + col;
                float v = alpha * acc[i][j][r];
                if (beta != 0.0f) v += beta * Cin[idx];
                D[idx] = v;
            }
        }
}

// ---------------------------------------------------------------------------
// Fused stage 2+3: out[o,m,n] = bias[o] + sum_{i,j} theta[i,j,o] *
//                               ( x_row[i] @ Tc[j] )[m,n]
// One wave owns one 16x16 (m,n) tile with 25 f32 WMMA accumulators (200
// VGPRs), then contracts with theta (staged in LDS) into 32 output channels.
// __launch_bounds__(256, 1): allow occupancy-1 so the 25 accumulators stay
// in VGPRs (gfx1250 allows up to 1024 VGPRs/wave) instead of spilling.
// ---------------------------------------------------------------------------
__global__ __launch_bounds__(256, 1)
void fused_stage23(const float* __restrict__ Af,    // NP x M x N (x_row)
                   const float* __restrict__ Tc,    // NP x N x N
                   const float* __restrict__ theta, // NP*NP*OUTC
                   const float* __restrict__ bias,  // OUTC
                   float* __restrict__ out) {       // OUTC x M x N
    __shared__ float sTheta[NP * NP * OUTC];
    __shared__ float sBias[OUTC];
    for (int t = threadIdx.x; t < NP * NP * OUTC; t += blockDim.x)
        sTheta[t] = theta[t];
    if (threadIdx.x < OUTC) sBias[threadIdx.x] = bias[threadIdx.x];
    __syncthreads();

    const size_t MN = (size_t)MDIM * NDIM;
    const int lane  = threadIdx.x & 31;
    const int wave  = blockIdx.x * (blockDim.x >> 5) + (threadIdx.x >> 5);
    const int tilesN = NDIM / 16;             // 96
    const int tmB = (wave / tilesN) * 16;
    const int tnB = (wave % tilesN) * 16;

    const int mr = lane & 15;
    const int kb = (lane >> 4) << 1;

    v8f acc[NP][NP] = {};

    const float* aptr[NP];
    const float* bptr[NP];
#pragma unroll
    for (int i = 0; i < NP; ++i)
        aptr[i] = Af + (size_t)i * MN + (size_t)(tmB + mr) * NDIM + kb;
#pragma unroll
    for (int j = 0; j < NP; ++j)
        bptr[j] = Tc + (size_t)j * MN + (size_t)kb * NDIM + tnB + mr;

    for (int k = 0; k < NDIM; k += 4) {
        // B fragments first (5 live v2f), then consume each A fragment
        // immediately against all 5 B fragments.
        v2f bf[NP];
#pragma unroll
        for (int j = 0; j < NP; ++j) {
            const float* p = bptr[j] + (size_t)k * NDIM;
            bf[j].x = p[0];
            bf[j].y = p[NDIM];
        }
#pragma unroll
        for (int i = 0; i < NP; ++i) {
            v2f af = *(const v2f*)(aptr[i] + k);
#pragma unroll
            for (int j = 0; j < NP; ++j)
                acc[i][j] = wmma_f32(af, bf[j], acc[i][j]);
        }
    }

    // Theta contraction + bias + store (C/D layout: VGPR r -> rows r / r+8)
    const int rbase = tmB + ((lane >> 4) << 3);
    const int col   = tnB + mr;
    for (int o = 0; o < OUTC; ++o) {
        float v[8];
#pragma unroll
        for (int r = 0; r < 8; ++r) v[r] = sBias[o];
#pragma unroll
        for (int i = 0; i < NP; ++i)
#pragma unroll
            for (int j = 0; j < NP; ++j) {
                float th = sTheta[(i * NP + j) * OUTC + o];
#pragma unroll
                for (int r = 0; r < 8; ++r) v[r] += th * acc[i][j][r];
            }
        float* po = out + (size_t)o * MN + (size_t)rbase * NDIM + col;
#pragma unroll
        for (int r = 0; r < 8; ++r) po[(size_t)r * NDIM] = v[r];
    }
}

// ---------------------------------------------------------------------------
extern "C" void kernel_launch(void* const* d_in, const int* in_sizes, int n_in,
                              void* d_out, int out_size, void* d_ws, size_t ws_size,
                              hipStream_t stream) {
    const float* x     = (const float*)d_in[0];  // 1 x M x N
    const float* Lr    = (const float*)d_in[1];  // M x M
    const float* Lc    = (const float*)d_in[2];  // N x N
    const float* theta = (const float*)d_in[3];  // NP x NP x 1 x OUTC
    const float* bias  = (const float*)d_in[4];  // OUTC
    float* out = (float*)d_out;

    const size_t MM = (size_t)MDIM * MDIM;       // == M*N == N*N here
    float* Tr = (float*)d_ws;                    // NP * MM
    float* Tc = Tr + (size_t)NP * MM;            // NP * MM
    float* Af = Tc + (size_t)NP * MM;            // NP * MM  (x_row)

    const int eltBlocks = (int)((MM + 255) / 256);           // 9216
    init_cheb01<<<eltBlocks, 256, 0, stream>>>(Lr, Tr, MDIM);
    init_cheb01<<<eltBlocks, 256, 0, stream>>>(Lc, Tc, NDIM);
    copy_f32<<<eltBlocks, 256, 0, stream>>>(x, Af, MM);      // x_row[0] = x

    // GEMM grid: (M/32)*(N/32) waves, 8 waves/block
    const int gemmBlocks = (MDIM / 32) * (NDIM / 32) / 8;    // 288

    // Chebyshev recursion: T_k = 2 * L @ T_{k-1} - T_{k-2}
    for (int k = 2; k < NP; ++k) {
        gemm32_f32<<<dim3(gemmBlocks, 1, 1), 256, 0, stream>>>(
            Lr, Tr + (size_t)(k - 1) * MM, Tr + (size_t)(k - 2) * MM,
            Tr + (size_t)k * MM, MDIM, MDIM, MDIM, 2.0f, -1.0f, 0, 0, 0);
        gemm32_f32<<<dim3(gemmBlocks, 1, 1), 256, 0, stream>>>(
            Lc, Tc + (size_t)(k - 1) * MM, Tc + (size_t)(k - 2) * MM,
            Tc + (size_t)k * MM, NDIM, NDIM, NDIM, 2.0f, -1.0f, 0, 0, 0);
    }

    // Stage 1 (batched over i = 1..4): x_row[i] = Tr[i] @ x
    gemm32_f32<<<dim3(gemmBlocks, 1, 4), 256, 0, stream>>>(
        Tr + MM, x, nullptr, Af + MM, MDIM, NDIM, MDIM, 1.0f, 0.0f,
        (long long)MM, 0, (long long)MM);

    // Fused stage 2+3
    const int fusedBlocks = (MDIM / 16) * (NDIM / 16) / 8;   // 1152
    fused_stage23<<<fusedBlocks, 256, 0, stream>>>(Af, Tc, theta, bias, out);
}